// myDotGATConv_16295105921120
// MI455X (gfx1250) — compile-verified
//
#include <hip/hip_runtime.h>
#include <math.h>

typedef __attribute__((ext_vector_type(2))) float v2f;
typedef __attribute__((ext_vector_type(8))) float v8f;

// Types for the async global->LDS builtin: param 0 is
// '__attribute__((vector_size(4*sizeof(int)))) int __device__ *' per hipcc.
typedef int async_v4i __attribute__((vector_size(16)));
typedef __attribute__((address_space(1))) async_v4i* async_gp;
typedef __attribute__((address_space(3))) async_v4i* async_lp;

#define HD 128           // num_heads * out_feats == in_feats == edge_feats
#define LDS_STRIDE 132   // 128 + 4 padding to dodge bank conflicts

__device__ __forceinline__ v8f wmma_f32x4(v2f a, v2f b, v8f c) {
    // 8 args: (neg_a, A, neg_b, B, c_mod, C, reuse_a, reuse_b)
    return __builtin_amdgcn_wmma_f32_16x16x4_f32(false, a, false, b, (short)0, c,
                                                 false, false);
}

// Stage a 16 x 128 fp32 tile (rows clamped) into LDS with padded stride.
// Uses gfx1250 async global->LDS B128 transfers when the builtin is available
// (ASYNCcnt-tracked, no VGPR round trip); falls back to regular loads.
__device__ __forceinline__ void stage_tile(const float* __restrict__ A, int row0,
                                           int rows, float* lds, int tid) {
    for (int i = tid; i < 16 * 32; i += 256) {        // 512 float4 slots
        int r  = i >> 5;                              // 0..15
        int c4 = i & 31;                              // float4 column
        int row = row0 + r;
        float* p = lds + r * LDS_STRIDE + c4 * 4;
#if __has_builtin(__builtin_amdgcn_global_load_async_to_lds_b128)
        if (row < rows) {
            __builtin_amdgcn_global_load_async_to_lds_b128(
                (async_gp)(A + (size_t)row * HD + c4 * 4),
                (async_lp)p, 0, 0);
        } else {
            p[0] = 0.f; p[1] = 0.f; p[2] = 0.f; p[3] = 0.f;
        }
#else
        float4 val = make_float4(0.f, 0.f, 0.f, 0.f);
        if (row < rows)
            val = ((const float4*)(A + (size_t)row * HD))[c4];
        p[0] = val.x; p[1] = val.y; p[2] = val.z; p[3] = val.w;
#endif
    }
#if __has_builtin(__builtin_amdgcn_global_load_async_to_lds_b128)
  #if __has_builtin(__builtin_amdgcn_s_wait_asynccnt)
    __builtin_amdgcn_s_wait_asynccnt(0);
  #else
    asm volatile("s_wait_asynccnt 0x0" ::: "memory");
  #endif
#endif
}

// ---------------------------------------------------------------------------
// Kernel 1: out = x (residual preload); m = -inf; den = 0
// ---------------------------------------------------------------------------
__global__ void init_kernel(const float* __restrict__ x, float* __restrict__ out,
                            float* __restrict__ m, float* __restrict__ den,
                            int nfeat_total, int nh_total) {
    int i = blockIdx.x * blockDim.x + threadIdx.x;
    if (i < nfeat_total) out[i] = x[i];
    if (i < nh_total) { m[i] = -INFINITY; den[i] = 0.0f; }
}

// ---------------------------------------------------------------------------
// Kernel 2: fused q/k/v projection.  Block = 256 thr = 8 waves.
// Block handles 16 rows; wave w handles output columns [16w, 16w+16).
// ---------------------------------------------------------------------------
__global__ __launch_bounds__(256) void proj3_kernel(
    const float* __restrict__ x,
    const float* __restrict__ Wq, const float* __restrict__ bq,
    const float* __restrict__ Wk, const float* __restrict__ bk,
    const float* __restrict__ Wv, const float* __restrict__ bv,
    float* __restrict__ q, float* __restrict__ k, float* __restrict__ v,
    int rows) {
    __shared__ float a_lds[16 * LDS_STRIDE];
    const int tid  = threadIdx.x;
    const int wave = tid >> 5;
    const int lane = tid & 31;
    const int row0 = blockIdx.x * 16;

    stage_tile(x, row0, rows, a_lds, tid);
    __syncthreads();

    const int col0 = wave * 16;
    const int half = lane >> 4;   // 0: K pair {0,1} / rows 0..7 ; 1: {2,3} / rows 8..15
    const int l15  = lane & 15;

    v8f accq = {}; v8f acck = {}; v8f accv = {};
    for (int kk = 0; kk < HD; kk += 4) {
        v2f af;
        af.x = a_lds[l15 * LDS_STRIDE + kk + half * 2 + 0];
        af.y = a_lds[l15 * LDS_STRIDE + kk + half * 2 + 1];
        const int r0 = (kk + half * 2 + 0) * HD + col0 + l15;
        const int r1 = (kk + half * 2 + 1) * HD + col0 + l15;
        v2f bfq; bfq.x = Wq[r0]; bfq.y = Wq[r1];
        v2f bfk; bfk.x = Wk[r0]; bfk.y = Wk[r1];
        v2f bfv; bfv.x = Wv[r0]; bfv.y = Wv[r1];
        accq = wmma_f32x4(af, bfq, accq);
        acck = wmma_f32x4(af, bfk, acck);
        accv = wmma_f32x4(af, bfv, accv);
    }

    const float biq = bq[col0 + l15];
    const float bik = bk[col0 + l15];
    const float biv = bv[col0 + l15];
    #pragma unroll
    for (int r = 0; r < 8; ++r) {
        int row = row0 + r + half * 8;
        if (row < rows) {
            size_t o = (size_t)row * HD + col0 + l15;
            q[o] = accq[r] + biq;
            k[o] = acck[r] + bik;
            v[o] = accv[r] + biv;
        }
    }
}

// ---------------------------------------------------------------------------
// Kernel 3: fused edge projection (WMMA -> LDS) + attention logit.
// Block handles 16 edges.  Phase 1: e = efeat@We + be into LDS.
// Phase 2: a[e,h] = <q[dst], k[src] + e> / sqrt(32); nan-flag -> 1e-9.
// ---------------------------------------------------------------------------
__global__ __launch_bounds__(256) void edge_logit_kernel(
    const float* __restrict__ efeat,
    const float* __restrict__ We, const float* __restrict__ be,
    const float* __restrict__ q, const float* __restrict__ k,
    const int* __restrict__ src, const int* __restrict__ dst,
    const int* __restrict__ nanf,
    float* __restrict__ a_out, int nedges) {
    __shared__ float a_lds[16 * LDS_STRIDE];
    __shared__ float e_lds[16 * LDS_STRIDE];
    const int tid  = threadIdx.x;
    const int wave = tid >> 5;
    const int lane = tid & 31;
    const int row0 = blockIdx.x * 16;

    stage_tile(efeat, row0, nedges, a_lds, tid);
    __syncthreads();

    const int col0 = wave * 16;
    const int half = lane >> 4;
    const int l15  = lane & 15;

    v8f acc = {};
    for (int kk = 0; kk < HD; kk += 4) {
        v2f af;
        af.x = a_lds[l15 * LDS_STRIDE + kk + half * 2 + 0];
        af.y = a_lds[l15 * LDS_STRIDE + kk + half * 2 + 1];
        v2f bf;
        bf.x = We[(kk + half * 2 + 0) * HD + col0 + l15];
        bf.y = We[(kk + half * 2 + 1) * HD + col0 + l15];
        acc = wmma_f32x4(af, bf, acc);
    }
    const float bie = be[col0 + l15];
    #pragma unroll
    for (int r = 0; r < 8; ++r)
        e_lds[(r + half * 8) * LDS_STRIDE + col0 + l15] = acc[r] + bie;
    __syncthreads();

    // Phase 2: 64 (edge,head) pairs x 4 threads each, 8 elements per thread.
    int pair = tid >> 2;            // 0..63
    int sub  = tid & 3;             // 0..3  (== lane & 3, contiguous in wave)
    if (pair < 64) {
        int eloc = pair & 15;
        int head = pair >> 4;       // 0..3
        int eg   = row0 + eloc;
        if (eg < nedges) {
            int s = src[eg], d = dst[eg];
            int base = head * 32 + sub * 8;
            const float* qp = q + (size_t)d * HD + base;
            const float* kp = k + (size_t)s * HD + base;
            const float* ep = &e_lds[eloc * LDS_STRIDE + base];
            float sum = 0.f;
            #pragma unroll
            for (int i = 0; i < 8; ++i) sum += qp[i] * (kp[i] + ep[i]);
            sum += __shfl_xor(sum, 1, 32);
            sum += __shfl_xor(sum, 2, 32);
            float aval = sum * 0.17677669529663687f;   // 1/sqrt(32)
            if (nanf[eg] != 0) aval = 1e-9f;
            if (sub == 0) a_out[(size_t)eg * 4 + head] = aval;
        }
    }
}

// ---------------------------------------------------------------------------
// Kernel 4: segmented max over destination nodes (float atomic max via bits).
// ---------------------------------------------------------------------------
__device__ __forceinline__ void atomicMaxFloat(float* addr, float val) {
    if (val >= 0.0f)
        atomicMax((int*)addr, __float_as_int(val));
    else
        atomicMin((unsigned int*)addr, __float_as_uint(val));
}

__global__ void segmax_kernel(const float* __restrict__ a, const int* __restrict__ dst,
                              float* __restrict__ m, int total) {
    int idx = blockIdx.x * blockDim.x + threadIdx.x;
    if (idx >= total) return;
    int e = idx >> 2, h = idx & 3;
    atomicMaxFloat(&m[dst[e] * 4 + h], a[idx]);
}

// ---------------------------------------------------------------------------
// Kernel 5: ex = exp(a - m[dst]) (in place), den[dst] += ex.
// ---------------------------------------------------------------------------
__global__ void expsum_kernel(float* __restrict__ a, const int* __restrict__ dst,
                              const float* __restrict__ m, float* __restrict__ den,
                              int total) {
    int idx = blockIdx.x * blockDim.x + threadIdx.x;
    if (idx >= total) return;
    int e = idx >> 2, h = idx & 3;
    int d = dst[e];
    float ex = expf(a[idx] - m[d * 4 + h]);
    a[idx] = ex;
    atomicAdd(&den[d * 4 + h], ex);
}

// ---------------------------------------------------------------------------
// Kernel 6: out[dst] += (ex/den[dst]) * v[src]; one wave per edge.
// ---------------------------------------------------------------------------
__global__ __launch_bounds__(256) void agg_kernel(
    const float* __restrict__ ex, const float* __restrict__ den,
    const float* __restrict__ v, const int* __restrict__ src,
    const int* __restrict__ dst, const int* __restrict__ nanf,
    float* __restrict__ out, int nedges) {
    int gtid = blockIdx.x * 256 + threadIdx.x;
    int e    = gtid >> 5;
    int lane = gtid & 31;
    if (e >= nedges) return;
    if (nanf[e] != 0) return;          // sa forced to 0 after softmax
    int s = src[e], d = dst[e];
    #pragma unroll
    for (int h = 0; h < 4; ++h) {
        float sa = ex[(size_t)e * 4 + h] / den[d * 4 + h];
        int c = h * 32 + lane;
        atomicAdd(&out[(size_t)d * HD + c], sa * v[(size_t)s * HD + c]);
    }
}

// ---------------------------------------------------------------------------
extern "C" void kernel_launch(void* const* d_in, const int* in_sizes, int n_in,
                              void* d_out, int out_size, void* d_ws, size_t ws_size,
                              hipStream_t stream) {
    const float* x     = (const float*)d_in[0];
    const float* efeat = (const float*)d_in[1];
    const int*   src   = (const int*)d_in[2];
    const int*   dst   = (const int*)d_in[3];
    const int*   nanf  = (const int*)d_in[4];
    const float* Wq    = (const float*)d_in[5];
    const float* bq    = (const float*)d_in[6];
    const float* Wk    = (const float*)d_in[7];
    const float* bk    = (const float*)d_in[8];
    const float* Wv    = (const float*)d_in[9];
    const float* bv    = (const float*)d_in[10];
    const float* We    = (const float*)d_in[11];
    const float* be    = (const float*)d_in[12];
    float* out = (float*)d_out;

    const int N = in_sizes[0] / HD;        // 50000
    const int E = in_sizes[2];             // 800000
    const int NH = N * 4;

    float* ws  = (float*)d_ws;
    float* q   = ws;                       // N*128
    float* k   = q + (size_t)N * HD;       // N*128
    float* v   = k + (size_t)N * HD;       // N*128
    float* a   = v + (size_t)N * HD;       // E*4 (logits, then exp in place)
    float* m   = a + (size_t)E * 4;        // N*4
    float* den = m + (size_t)NH;           // N*4

    // 1. init: out = x, m = -inf, den = 0
    {
        int total = N * HD;
        init_kernel<<<(total + 255) / 256, 256, 0, stream>>>(x, out, m, den, total, NH);
    }
    // 2. q/k/v projections (fused, WMMA f32)
    proj3_kernel<<<(N + 15) / 16, 256, 0, stream>>>(x, Wq, bq, Wk, bk, Wv, bv,
                                                    q, k, v, N);
    // 3. edge projection + logits (fused, WMMA f32)
    edge_logit_kernel<<<(E + 15) / 16, 256, 0, stream>>>(efeat, We, be, q, k,
                                                         src, dst, nanf, a, E);
    // 4. segment max
    segmax_kernel<<<(E * 4 + 255) / 256, 256, 0, stream>>>(a, dst, m, E * 4);
    // 5. exp + segment sum
    expsum_kernel<<<(E * 4 + 255) / 256, 256, 0, stream>>>(a, dst, m, den, E * 4);
    // 6. weighted aggregation (one wave per edge)
    {
        long long threads = (long long)E * 32;
        agg_kernel<<<(int)((threads + 255) / 256), 256, 0, stream>>>(a, den, v, src,
                                                                     dst, nanf, out, E);
    }
}